// GNNLayer_24730421690403
// MI455X (gfx1250) — compile-verified
//
#include <hip/hip_runtime.h>

#define H 512

typedef __attribute__((ext_vector_type(2))) float v2f;
typedef __attribute__((ext_vector_type(8))) float v8f;

// ---------------------------------------------------------------------------
// Kernel 1: zero the per-node edge counters.
// ---------------------------------------------------------------------------
__global__ void gnn_zero_kernel(int* __restrict__ cnt, int n) {
    int i = blockIdx.x * blockDim.x + threadIdx.x;
    if (i < n) cnt[i] = 0;
}

// ---------------------------------------------------------------------------
// Kernel 2: histogram destinations (800K int atomics, not 400M f32 atomics).
// ---------------------------------------------------------------------------
__global__ void gnn_count_kernel(int* __restrict__ cnt, const int* __restrict__ ei, int E) {
    int e = blockIdx.x * blockDim.x + threadIdx.x;
    if (e < E) atomicAdd(&cnt[ei[E + e]], 1);
}

// ---------------------------------------------------------------------------
// Kernel 3: exclusive prefix sum over 50000 counts.
// Single 1024-thread block (32 waves), LDS Hillis-Steele per chunk + carry.
// ---------------------------------------------------------------------------
__global__ void __launch_bounds__(1024)
gnn_scan_kernel(const int* __restrict__ cnt, int* __restrict__ start,
                int* __restrict__ cursor, int N) {
    __shared__ int buf[1024];
    __shared__ int carry;
    const int t = threadIdx.x;
    if (t == 0) carry = 0;
    __syncthreads();
    for (int base = 0; base < N; base += 1024) {
        const int i = base + t;
        const int v = (i < N) ? cnt[i] : 0;
        buf[t] = v;
        __syncthreads();
        #pragma unroll
        for (int off = 1; off < 1024; off <<= 1) {
            int add = (t >= off) ? buf[t - off] : 0;
            __syncthreads();
            buf[t] += add;
            __syncthreads();
        }
        const int incl = buf[t];
        const int excl = incl - v;
        if (i < N) {
            start[i]  = carry + excl;
            cursor[i] = carry + excl;
        }
        __syncthreads();                 // everyone has read carry
        if (t == 1023) carry += incl;    // chunk total
        __syncthreads();
    }
    if (t == 0) start[N] = carry;        // == E
}

// ---------------------------------------------------------------------------
// Kernel 4: bin edge sources by destination (counting-sort scatter).
// ---------------------------------------------------------------------------
__global__ void gnn_fill_kernel(int* __restrict__ bins, int* __restrict__ cursor,
                                const int* __restrict__ ei, int E) {
    int e = blockIdx.x * blockDim.x + threadIdx.x;
    if (e < E) {
        int pos = atomicAdd(&cursor[ei[E + e]], 1);
        bins[pos] = ei[e];               // src
    }
}

// ---------------------------------------------------------------------------
// Kernel 4b: sort each bin by source index (insertion sort, one thread/node).
// Makes the fp summation order a pure function of the inputs, restoring
// determinism that the atomic cursor allocation would otherwise break.
// bins is 3.2MB -> L2 resident; avg degree 16, so ~128 L2 accesses/thread.
// ---------------------------------------------------------------------------
__global__ void gnn_sort_kernel(int* __restrict__ bins, const int* __restrict__ start, int N) {
    int n = blockIdx.x * blockDim.x + threadIdx.x;
    if (n >= N) return;
    const int s0 = start[n];
    const int s1 = start[n + 1];
    for (int i = s0 + 1; i < s1; ++i) {
        const int key = bins[i];
        int j = i - 1;
        while (j >= s0 && bins[j] > key) {
            bins[j + 1] = bins[j];
            --j;
        }
        bins[j + 1] = key;
    }
}

// ---------------------------------------------------------------------------
// Kernel 5: h[n] = x[n] + sum_{e in bin(n)} x[src(e)].
// 2 nodes per 256-thread block; 128 threads cover the 512-float row as float4.
// Unrolled x4: batch index loads, then 4 independent row loads in flight;
// two accumulator chains break the FP add dependency.
// ---------------------------------------------------------------------------
__global__ void gnn_gather_kernel(float* __restrict__ h, const float* __restrict__ x,
                                  const int* __restrict__ bins,
                                  const int* __restrict__ start, int N) {
    const int n = blockIdx.x * 2 + (threadIdx.x >> 7);
    if (n >= N) return;
    const int f  = (threadIdx.x & 127) << 2;     // 0..508 step 4
    const int s0 = start[n];
    const int s1 = start[n + 1];

    float4 acc0 = *(const float4*)(x + (size_t)n * H + f);
    float4 acc1 = {0.f, 0.f, 0.f, 0.f};

    int i = s0;
    for (; i + 4 <= s1; i += 4) {
        const int i0 = bins[i + 0];
        const int i1 = bins[i + 1];
        const int i2 = bins[i + 2];
        const int i3 = bins[i + 3];
        const float4 v0 = *(const float4*)(x + (size_t)i0 * H + f);
        const float4 v1 = *(const float4*)(x + (size_t)i1 * H + f);
        const float4 v2 = *(const float4*)(x + (size_t)i2 * H + f);
        const float4 v3 = *(const float4*)(x + (size_t)i3 * H + f);
        acc0.x += v0.x; acc0.y += v0.y; acc0.z += v0.z; acc0.w += v0.w;
        acc1.x += v1.x; acc1.y += v1.y; acc1.z += v1.z; acc1.w += v1.w;
        acc0.x += v2.x; acc0.y += v2.y; acc0.z += v2.z; acc0.w += v2.w;
        acc1.x += v3.x; acc1.y += v3.y; acc1.z += v3.z; acc1.w += v3.w;
    }
    for (; i < s1; ++i) {
        const int src = bins[i];
        const float4 v = *(const float4*)(x + (size_t)src * H + f);
        acc0.x += v.x; acc0.y += v.y; acc0.z += v.z; acc0.w += v.w;
    }
    acc0.x += acc1.x; acc0.y += acc1.y; acc0.z += acc1.z; acc0.w += acc1.w;
    *(float4*)(h + (size_t)n * H + f) = acc0;
}

// ---------------------------------------------------------------------------
// Kernel 6: out = relu(h @ W + b) using V_WMMA_F32_16X16X4_F32.
// Block: 256 threads = 8 waves; each wave 16 rows x 64 cols (4 accumulators).
// Block tile 128x64; all waves share B fragments (WGP$ reuse).
// ---------------------------------------------------------------------------
__global__ void __launch_bounds__(256)
gnn_gemm_relu_kernel(float* __restrict__ out, const float* __restrict__ h,
                     const float* __restrict__ W, const float* __restrict__ bias, int N) {
    const int lane = threadIdx.x & 31;
    const int wave = threadIdx.x >> 5;
    const int half = lane >> 4;                // 0: K=0,1 / M=r   1: K=2,3 / M=r+8
    const int l16  = lane & 15;

    const int rowBase = blockIdx.x * 128 + wave * 16;
    const int colBase = blockIdx.y * 64;

    // Clamp A rows (WMMA requires EXEC all-ones); mask only the final store.
    int arow = rowBase + l16;
    if (arow >= N) arow = N - 1;
    const float* hrow = h + (size_t)arow * H + 2 * half;

    v8f c0 = {}, c1 = {}, c2 = {}, c3 = {};

#pragma unroll 4
    for (int k = 0; k < H; k += 4) {
        v2f a = *(const v2f*)(hrow + k);

        const float* wp = W + (size_t)(k + 2 * half) * H + colBase + l16;
        v2f b0, b1, b2, b3;
        b0.x = wp[0];      b0.y = wp[H];
        b1.x = wp[16];     b1.y = wp[H + 16];
        b2.x = wp[32];     b2.y = wp[H + 32];
        b3.x = wp[48];     b3.y = wp[H + 48];

        c0 = __builtin_amdgcn_wmma_f32_16x16x4_f32(false, a, false, b0, (short)0, c0, false, false);
        c1 = __builtin_amdgcn_wmma_f32_16x16x4_f32(false, a, false, b1, (short)0, c1, false, false);
        c2 = __builtin_amdgcn_wmma_f32_16x16x4_f32(false, a, false, b2, (short)0, c2, false, false);
        c3 = __builtin_amdgcn_wmma_f32_16x16x4_f32(false, a, false, b3, (short)0, c3, false, false);
    }

    const int col0 = colBase + l16;
    const float bb0 = bias[col0];
    const float bb1 = bias[col0 + 16];
    const float bb2 = bias[col0 + 32];
    const float bb3 = bias[col0 + 48];

#pragma unroll
    for (int r = 0; r < 8; ++r) {
        const int row = rowBase + r + 8 * half;   // C/D: VGPR r -> M=r (+8 upper half)
        if (row < N) {
            float* op = out + (size_t)row * H + col0;
            float v0 = c0[r] + bb0; op[0]  = v0 > 0.f ? v0 : 0.f;
            float v1 = c1[r] + bb1; op[16] = v1 > 0.f ? v1 : 0.f;
            float v2 = c2[r] + bb2; op[32] = v2 > 0.f ? v2 : 0.f;
            float v3 = c3[r] + bb3; op[48] = v3 > 0.f ? v3 : 0.f;
        }
    }
}

// ---------------------------------------------------------------------------
extern "C" void kernel_launch(void* const* d_in, const int* in_sizes, int n_in,
                              void* d_out, int out_size, void* d_ws, size_t ws_size,
                              hipStream_t stream) {
    const float* x    = (const float*)d_in[0];   // [N, 512] f32
    const int*   ei   = (const int*)d_in[1];     // [2, E] int
    const float* W    = (const float*)d_in[2];   // [512, 512] f32
    const float* bias = (const float*)d_in[3];   // [512] f32
    float* out = (float*)d_out;                  // [N, 512] f32

    const int N = in_sizes[0] / H;               // 50000
    const int E = in_sizes[1] / 2;               // 800000

    // Workspace layout: h | cnt | start | cursor | bins
    char* ws = (char*)d_ws;
    float* h      = (float*)ws;                       ws += (size_t)N * H * sizeof(float);
    int*   cnt    = (int*)ws;                         ws += (size_t)N * sizeof(int);
    int*   start  = (int*)ws;                         ws += (size_t)(N + 1) * sizeof(int);
    int*   cursor = (int*)ws;                         ws += (size_t)N * sizeof(int);
    int*   bins   = (int*)ws;

    // 1) cnt = 0
    gnn_zero_kernel<<<(N + 255) / 256, 256, 0, stream>>>(cnt, N);
    // 2) cnt[dst]++
    gnn_count_kernel<<<(E + 255) / 256, 256, 0, stream>>>(cnt, ei, E);
    // 3) start/cursor = exclusive_scan(cnt)
    gnn_scan_kernel<<<1, 1024, 0, stream>>>(cnt, start, cursor, N);
    // 4) bins[cursor[dst]++] = src
    gnn_fill_kernel<<<(E + 255) / 256, 256, 0, stream>>>(bins, cursor, ei, E);
    // 4b) sort each bin -> deterministic summation order
    gnn_sort_kernel<<<(N + 255) / 256, 256, 0, stream>>>(bins, start, N);
    // 5) h[n] = x[n] + sum of x over bin(n)
    gnn_gather_kernel<<<(N + 1) / 2, 256, 0, stream>>>(h, x, bins, start, N);
    // 6) out = relu(h @ W + b)
    dim3 grid((N + 127) / 128, H / 64);
    gnn_gemm_relu_kernel<<<grid, 256, 0, stream>>>(out, h, W, bias, N);
}